// PCSM_Loss_51342039056871
// MI455X (gfx1250) — compile-verified
//
#include <hip/hip_runtime.h>
#include <math.h>

// ---------------------------------------------------------------------------
// PCSM loss on MI455X (gfx1250, wave32), fused single-pass version.
//  Stage 0: zero Gram/avg accumulators (8.4 KB in d_ws)
//  Stage 1: fused kernel per image:
//           global -> OD transcendental math -> LDS 32x128 FOD tile
//           -> ds_load_b64 -> V_WMMA_F32_16X16X4_F32 Gram accumulation
//           -> LDS ds_add_f32 wave-combine -> global atomics
//  Stage 2: cosine-Gram L1 + avg MSE -> scalar
// Traffic: only the 2x100.7MB image reads (+ tiny atomics) ~ 9us @ 23.3TB/s.
// ---------------------------------------------------------------------------

typedef __attribute__((ext_vector_type(2))) float v2f;
typedef __attribute__((ext_vector_type(8))) float v8f;

#define NB     32           // batch
#define HWPX   262144       // 512*512 pixels
#define CHUNK  128          // pixels per LDS tile
#define NCHUNK (HWPX / CHUNK)       // 2048
#define LSTR   132          // LDS row stride (pad: 4*r mod 64 distinct banks)
// HED_FROM_RGB[:,2] / LOG_ADJUST  (LOG_ADJUST = ln(1e-6) = -13.8155105579)
#define C_R   0.04025303f
#define C_G   0.00980938f
#define C_B  -0.11390010f
// LOG_ADJUST * RGB_FROM_HED[2] = -13.8155106 * {0.27, 0.57, 0.78}
#define K_R  -3.73018785f
#define K_G  -7.87484102f
#define K_B  -10.77609824f
#define ADJUST_CAL 0.01807546f   // 10^(-e^(1/1.8))
#define ALPHA 1.8f
#define THRESH_FOD 0.15f
#define COS_EPS 1e-8f

__device__ __forceinline__ float fod_of_pixel(float r, float g, float b,
                                              float* fod_relu) {
    r = fmaxf(r, 1e-6f); g = fmaxf(g, 1e-6f); b = fmaxf(b, 1e-6f);
    // DAB channel of separate_stains (only column 2 of HED_FROM_RGB needed)
    float d = fmaxf(__logf(r) * C_R + __logf(g) * C_G + __logf(b) * C_B, 0.0f);
    // combine_stains with only DAB: exp(LOG_ADJUST * d * RGB_FROM_HED[2])
    float e0 = __expf(K_R * d);   // d>=0, K<0 -> e in (0,1], clip is a no-op
    float e1 = __expf(K_G * d);
    float e2 = __expf(K_B * d);
    float grey = fminf(0.2125f * e0 + 0.7154f * e1 + 0.0721f * e2, 1.0f);
    float t = fmaxf(-__log10f(grey + ADJUST_CAL), 0.0f);
    float fod = (t > 0.0f) ? __expf(ALPHA * __logf(t)) : 0.0f;  // t^1.8
    *fod_relu = (fod < THRESH_FOD) ? 0.0f : fod;
    return fod;
}

// ---------------------------------------------------------------------------
__global__ void zero_acc_kernel(float* __restrict__ acc, int n) {
    int t = blockIdx.x * blockDim.x + threadIdx.x;
    if (t < n) acc[t] = 0.0f;
}

// ---------------------------------------------------------------------------
// Fused OD + Gram. Block = 256 threads = 8 waves.
// Per chunk of 128 pixels: every thread computes FOD for 16 pixels of batch
// row b = t>>3 (float4 coalesced across the 8 threads sharing one row),
// stores to LDS tile[32][LSTR]; then each wave runs 4 K-steps of
// V_WMMA_F32_16X16X4_F32 over its 16-pixel K-slice, accumulating the full
// 32x32 Gram in four 16x16 register tiles across all chunks.
// A/B fragments are identical layouts: lane L holds tile[base+(L&15)][k+2*(L>>4)..+1].
// ---------------------------------------------------------------------------
__global__ void __launch_bounds__(256)
pcsm_fused_kernel(const float* __restrict__ img,
                  float* __restrict__ S,
                  float* __restrict__ avg) {
    __shared__ float tile[NB * LSTR];          // 16.9 KB

    const int t    = threadIdx.x;
    const int lane = t & 31;
    const int wave = t >> 5;
    const int b    = t >> 3;                   // batch row this thread produces
    const int s8   = t & 7;                    // position among 8 threads/row

    // WMMA operand base addresses in LDS (8-byte aligned -> ds_load_b64)
    const int r     = lane & 15;
    const int khalf = (lane >> 4) << 1;        // 0 or 2
    const float* lp0 = &tile[r * LSTR + khalf];          // batches 0..15
    const float* lp1 = &tile[(r + 16) * LSTR + khalf];   // batches 16..31

    v8f c00 = {}, c01 = {}, c10 = {}, c11 = {};
    float avg_acc = 0.0f;

    for (int chunk = blockIdx.x; chunk < NCHUNK; chunk += gridDim.x) {
        const long n0 = (long)chunk * CHUNK;

        #pragma unroll
        for (int q = 0; q < 4; ++q) {
            const int  off  = s8 * 4 + q * 32;             // 0..124, 4 pixels
            const long base = (long)b * 3 * HWPX + n0 + off;
            float4 rr = *(const float4*)(img + base);
            float4 gg = *(const float4*)(img + base + HWPX);
            float4 bb = *(const float4*)(img + base + 2 * HWPX);

            if (q == 0 && chunk + gridDim.x < NCHUNK) {    // stream next chunk
                const long nb = base + (long)gridDim.x * CHUNK;
                __builtin_prefetch(img + nb, 0, 1);
                __builtin_prefetch(img + nb + HWPX, 0, 1);
                __builtin_prefetch(img + nb + 2 * HWPX, 0, 1);
            }

            float f0, f1, f2, f3;
            float4 f;
            f.x = fod_of_pixel(rr.x, gg.x, bb.x, &f0);
            f.y = fod_of_pixel(rr.y, gg.y, bb.y, &f1);
            f.z = fod_of_pixel(rr.z, gg.z, bb.z, &f2);
            f.w = fod_of_pixel(rr.w, gg.w, bb.w, &f3);
            avg_acc += (f0 + f1) + (f2 + f3);
            *(float4*)&tile[b * LSTR + off] = f;           // ds_store_b128
        }
        __syncthreads();

        // this wave's K-slice: pixels [wave*16, wave*16+16) of the chunk
        const int kb = wave * 16;
        #pragma unroll
        for (int kk = 0; kk < 16; kk += 4) {
            v2f a0 = *(const v2f*)(lp0 + kb + kk);
            v2f a1 = *(const v2f*)(lp1 + kb + kk);
            c00 = __builtin_amdgcn_wmma_f32_16x16x4_f32(false, a0, false, a0,
                                                        (short)0, c00, false, false);
            c01 = __builtin_amdgcn_wmma_f32_16x16x4_f32(false, a0, false, a1,
                                                        (short)0, c01, false, false);
            c10 = __builtin_amdgcn_wmma_f32_16x16x4_f32(false, a1, false, a0,
                                                        (short)0, c10, false, false);
            c11 = __builtin_amdgcn_wmma_f32_16x16x4_f32(false, a1, false, a1,
                                                        (short)0, c11, false, false);
        }
        __syncthreads();
    }

    // ---- per-batch avg: 8 threads own row b -> width-8 shuffle reduction
    float s = avg_acc;
    s += __shfl_down(s, 4, 8);
    s += __shfl_down(s, 2, 8);
    s += __shfl_down(s, 1, 8);
    if (s8 == 0) atomicAdd(&avg[b], s);

    // ---- Gram flush: combine the block's 8 waves in LDS, then one global
    //      atomic per element per block.
    __syncthreads();                       // tile free for reuse
    float* gacc = tile;                    // reuse first 1024 floats
    for (int i = t; i < 1024; i += 256) gacc[i] = 0.0f;
    __syncthreads();

    // C/D layout: VGPR v, lanes 0-15 -> (M=v, N=lane); lanes 16-31 -> M=8+v
    const int mrow = (lane >> 4) * 8;
    const int col  = lane & 15;
    #pragma unroll
    for (int v = 0; v < 8; ++v) {
        const int m = mrow + v;
        atomicAdd(&gacc[m * 32 + col],              c00[v]);  // ds_add_f32
        atomicAdd(&gacc[m * 32 + 16 + col],         c01[v]);
        atomicAdd(&gacc[(16 + m) * 32 + col],       c10[v]);
        atomicAdd(&gacc[(16 + m) * 32 + 16 + col],  c11[v]);
    }
    __syncthreads();
    for (int i = t; i < 1024; i += 256) atomicAdd(&S[i], gacc[i]);
}

// ---------------------------------------------------------------------------
// cosine(S) from raw Grams: cos[i,j] = S[i,j]/(max(sqrt(S[ii]),eps)*max(sqrt(S[jj]),eps))
// ---------------------------------------------------------------------------
__global__ void finalize_kernel(const float* __restrict__ S_in,
                                const float* __restrict__ S_out,
                                const float* __restrict__ avg_in,
                                const float* __restrict__ avg_out,
                                float* __restrict__ out) {
    __shared__ float red[1024];
    const int t = threadIdx.x;          // 1024 threads = 32x32 matrix entries
    const int i = t >> 5, j = t & 31;

    float ni = fmaxf(sqrtf(S_in[i * 32 + i]), COS_EPS);
    float nj = fmaxf(sqrtf(S_in[j * 32 + j]), COS_EPS);
    float cin = S_in[i * 32 + j] / (ni * nj);
    float mi = fmaxf(sqrtf(S_out[i * 32 + i]), COS_EPS);
    float mj = fmaxf(sqrtf(S_out[j * 32 + j]), COS_EPS);
    float cout = S_out[i * 32 + j] / (mi * mj);

    red[t] = fabsf(cin - cout);
    __syncthreads();
    for (int sdx = 512; sdx > 0; sdx >>= 1) {
        if (t < sdx) red[t] += red[t + sdx];
        __syncthreads();
    }
    if (t == 0) {
        float l1 = red[0] * (1.0f / 1024.0f);
        float mse = 0.0f;
        #pragma unroll
        for (int bb = 0; bb < NB; ++bb) {
            float d = avg_in[bb] - avg_out[bb];
            mse += d * d;
        }
        mse = (mse * (1.0f / (float)NB)) / (262144.0f * 262144.0f);
        out[0] = l1 + mse;
    }
}

// ---------------------------------------------------------------------------
extern "C" void kernel_launch(void* const* d_in, const int* in_sizes, int n_in,
                              void* d_out, int out_size, void* d_ws, size_t ws_size,
                              hipStream_t stream) {
    const float* tgt = (const float*)d_in[1];   // inp image (src_matrix side)
    const float* gen = (const float*)d_in[2];   // out image (tgt_matrix side)

    float* acc     = (float*)d_ws;     // 2112 floats total
    float* S_in    = acc;              // 1024
    float* S_out   = acc + 1024;       // 1024
    float* avg_in  = acc + 2048;       // 32
    float* avg_out = acc + 2080;       // 32

    zero_acc_kernel<<<(2112 + 255) / 256, 256, 0, stream>>>(acc, 2112);

    // 512 blocks x 256 threads; each block grid-strides over 4 of 2048 chunks
    pcsm_fused_kernel<<<512, 256, 0, stream>>>(tgt, S_in, avg_in);
    pcsm_fused_kernel<<<512, 256, 0, stream>>>(gen, S_out, avg_out);

    finalize_kernel<<<1, 1024, 0, stream>>>(S_in, S_out, avg_in, avg_out,
                                            (float*)d_out);
}